// SlidingWindowAttention_55576876810887
// MI455X (gfx1250) — compile-verified
//
#include <hip/hip_runtime.h>
#include <math.h>

// Sliding-window causal attention (B=4, T=2048, D=768, H=12, hd=64, W=128), fp32.
// Matmuls on the CDNA5 fp32 matrix core (V_WMMA_F32_16X16X4_F32); LDS staging
// through the CDNA5 async path (GLOBAL_LOAD_ASYNC_TO_LDS_B128 + ASYNCcnt) with
// double-buffered software pipelining in the GEMMs.
// Roofline: ~42 GFLOP vs ~160MB HBM traffic @23.3TB/s -> near memory bound even
// at fp32 matrix rates, so we keep reference precision (no bf16 downcast).

typedef __attribute__((ext_vector_type(2))) float v2f;
typedef __attribute__((ext_vector_type(8))) float v8f;

static constexpr int kB   = 4;
static constexpr int kT   = 2048;
static constexpr int kD   = 768;
static constexpr int kH   = 12;
static constexpr int kHD  = 64;
static constexpr int kWIN = 128;
static constexpr int kDq  = 3 * kD;   // 2304
static constexpr int kM   = kB * kT;  // 8192

__device__ __forceinline__ v8f wmma4_f32(v2f a, v2f b, v8f c) {
  // D = A(16x4,f32) * B(4x16,f32) + C(16x16,f32)
  return __builtin_amdgcn_wmma_f32_16x16x4_f32(false, a, false, b, (short)0, c,
                                               false, false);
}

__device__ __forceinline__ v8f v8f_zero() {
  v8f z;
#pragma unroll
  for (int e = 0; e < 8; ++e) z[e] = 0.0f;
  return z;
}

// Async copy 16B/lane from global to LDS (VDST = per-lane LDS byte offset,
// which for a __shared__-derived generic pointer is its low 32 bits).
__device__ __forceinline__ void async_ld16(const void* g, void* l) {
  asm volatile("global_load_async_to_lds_b128 %0, %1, off"
               :
               : "v"((unsigned)(uintptr_t)l), "v"(g)
               : "memory");
}
#define ASYNC_WAIT_6() asm volatile("s_wait_asynccnt 0x6" ::: "memory")
#define ASYNC_WAIT_0() asm volatile("s_wait_asynccnt 0x0" ::: "memory")

// ---------------------------------------------------------------------------
// Generic fp32 WMMA GEMM: C[M,N] = A[M,K] * B[K,N], row-major.
// Block: 256 threads = 8 waves (4 in M x 2 in N). Block tile 128x64,
// wave tile 32x32 (2x2 C-frags). K staged in chunks of 32 through LDS,
// double-buffered via GLOBAL_LOAD_ASYNC_TO_LDS_B128 + s_wait_asynccnt so the
// fetch of stage k+1 overlaps the WMMAs of stage k. LDS strides 36 (A) /
// 72 (B) keep the WMMA frag reads bank-conflict free across wave32.
// ---------------------------------------------------------------------------
__global__ __launch_bounds__(256) void swa_gemm_f32_wmma(
    const float* __restrict__ A, const float* __restrict__ B,
    float* __restrict__ C, int M, int N, int K) {
  __shared__ float As[2][128 * 36];
  __shared__ float Bs[2][32 * 72];

  const int tid  = threadIdx.x;
  const int lane = tid & 31;
  const int wave = tid >> 5;   // 0..7
  const int wm   = wave >> 1;  // 0..3 (M)
  const int wn   = wave & 1;   // 0..1 (N)
  const int bm   = blockIdx.y;
  const int bn   = blockIdx.x;

  const int fm  = lane & 15;         // frag row (A) / col (B,C)
  const int fk0 = (lane >> 4) * 2;   // frag k start: 0 or 2
  const int fmb = (lane >> 4) * 8;   // C-frag row base

  // Per-thread staging coordinates (4 A-tiles + 2 B-tiles of 16B per thread).
  const int arow0 = tid >> 3;            // + i*32, i=0..3
  const int ac4   = (tid & 7) * 4;
  const int brow0 = tid >> 4;            // + i*16, i=0..1
  const int bc4   = (tid & 15) * 4;

  // Issue one K-stage of async loads into buffer `buf` (6 per wave).
  auto issueStage = [&](int kt, int buf) {
    const float* Ag = A + (size_t)(bm * 128) * K + kt * 32;
#pragma unroll
    for (int i = 0; i < 4; ++i) {
      int row = arow0 + i * 32;
      async_ld16(Ag + (size_t)row * K + ac4, &As[buf][row * 36 + ac4]);
    }
    const float* Bg = B + (size_t)(kt * 32) * N + bn * 64;
#pragma unroll
    for (int i = 0; i < 2; ++i) {
      int row = brow0 + i * 16;
      async_ld16(Bg + (size_t)row * N + bc4, &Bs[buf][row * 72 + bc4]);
    }
  };

  v8f acc[2][2];
#pragma unroll
  for (int i = 0; i < 2; ++i)
#pragma unroll
    for (int j = 0; j < 2; ++j) acc[i][j] = v8f_zero();

  const int kTiles = K >> 5;
  issueStage(0, 0);

  for (int kt = 0; kt < kTiles; ++kt) {
    const int buf = kt & 1;
    if (kt + 1 < kTiles) {
      issueStage(kt + 1, buf ^ 1);
      ASYNC_WAIT_6();  // stage kt's 6 loads retired (in-order), kt+1 in flight
    } else {
      ASYNC_WAIT_0();
    }
    __syncthreads();  // all waves' data for buf landed

#pragma unroll
    for (int ks = 0; ks < 8; ++ks) {  // 8 k-slices of 4
      v2f a[2], b[2];
#pragma unroll
      for (int mt = 0; mt < 2; ++mt) {
        const float* p = &As[buf][(wm * 32 + mt * 16 + fm) * 36 + ks * 4 + fk0];
        a[mt][0] = p[0];
        a[mt][1] = p[1];
      }
#pragma unroll
      for (int nt = 0; nt < 2; ++nt) {
        int col = wn * 32 + nt * 16 + fm;
        b[nt][0] = Bs[buf][(ks * 4 + fk0) * 72 + col];
        b[nt][1] = Bs[buf][(ks * 4 + fk0 + 1) * 72 + col];
      }
#pragma unroll
      for (int mt = 0; mt < 2; ++mt)
#pragma unroll
        for (int nt = 0; nt < 2; ++nt)
          acc[mt][nt] = wmma4_f32(a[mt], b[nt], acc[mt][nt]);
    }
    __syncthreads();  // done reading buf before it is refilled at kt+2
  }

#pragma unroll
  for (int mt = 0; mt < 2; ++mt)
#pragma unroll
    for (int nt = 0; nt < 2; ++nt) {
      int r0 = bm * 128 + wm * 32 + mt * 16 + fmb;
      int c0 = bn * 64 + wn * 32 + nt * 16 + fm;
#pragma unroll
      for (int r = 0; r < 8; ++r)
        C[(size_t)(r0 + r) * N + c0] = acc[mt][nt][r];
    }
}

// ---------------------------------------------------------------------------
// Sliding-window attention core. One block = 2 waves = 32 queries of one
// (b,h). Key window union: [q0-128, q0+31] = 160 rows staged in LDS through
// the async-to-LDS path. Wave w handles queries q0+16w..q0+16w+15; its
// 144-key band starts at LDS key-row 16w. Scores and P*V both via
// V_WMMA_F32_16X16X4_F32.
// ---------------------------------------------------------------------------
__global__ __launch_bounds__(64) void swa_attention_wmma(
    const float* __restrict__ qkv,  // [B*T, 3D]
    float* __restrict__ attn) {     // [B*T, D]
  __shared__ float Qs[32 * 68];       // 32 x 64, stride 68
  __shared__ float Ks[160 * 68];      // 160 x 64, stride 68
  __shared__ float Vs[160 * 72];      // 160 x 64, stride 72
  __shared__ float Ss[2 * 16 * 148];  // per-wave 16 x 144 scores, stride 148

  const int tid  = threadIdx.x;   // 0..63
  const int lane = tid & 31;
  const int w    = tid >> 5;      // wave 0..1

  const int idx  = blockIdx.x;          // b*H*(T/32) blocks
  const int qblk = idx & 63;            // T/32 = 64
  const int h    = (idx >> 6) % kH;
  const int b    = idx / (64 * kH);
  const int q0   = qblk * 32;
  const int kbase = q0 - kWIN;          // key j of LDS key-row 0

  const int fm  = lane & 15;
  const int fk0 = (lane >> 4) * 2;
  const int fmb = (lane >> 4) * 8;

  // ---- Stage Q / K / V into LDS (async 16B/lane; zero-fill j<0 rows) ----
  {
    const float* Qg = qkv + (size_t)(b * kT + q0) * kDq + h * kHD;
#pragma unroll
    for (int i = 0; i < 8; ++i) {  // 32 rows x 16 float4
      int pos = tid + i * 64;
      int row = pos >> 4;
      int c4  = (pos & 15) * 4;
      async_ld16(Qg + (size_t)row * kDq + c4, &Qs[row * 68 + c4]);
    }
    const float* Kg = qkv + (size_t)(b * kT) * kDq + kD + h * kHD;
    const float* Vg = qkv + (size_t)(b * kT) * kDq + 2 * kD + h * kHD;
    for (int i = 0; i < 40; ++i) {  // 160 rows x 16 float4
      int pos = tid + i * 64;
      int row = pos >> 4;
      int c4  = (pos & 15) * 4;
      int j   = kbase + row;
      if (j >= 0) {
        async_ld16(Kg + (size_t)j * kDq + c4, &Ks[row * 68 + c4]);
        async_ld16(Vg + (size_t)j * kDq + c4, &Vs[row * 72 + c4]);
      } else {
        float4 z = make_float4(0.f, 0.f, 0.f, 0.f);
        *(float4*)(&Ks[row * 68 + c4]) = z;
        *(float4*)(&Vs[row * 72 + c4]) = z;
      }
    }
  }
  ASYNC_WAIT_0();
  __syncthreads();

  const int w16 = w * 16;  // this wave's query/key-row offset

  // ---- Scores: S[16 x 144] = Q(16x64) * K^T, 9 key tiles x 16 k-slices ----
  for (int kt2 = 0; kt2 < 9; ++kt2) {
    v8f s = v8f_zero();
#pragma unroll
    for (int ks = 0; ks < 16; ++ks) {  // hd = 64 -> 16 slices of 4
      v2f a, bfr;
      const float* ap = &Qs[(w16 + fm) * 68 + ks * 4 + fk0];
      a[0] = ap[0];
      a[1] = ap[1];
      const float* bp = &Ks[(w16 + kt2 * 16 + fm) * 68 + ks * 4 + fk0];
      bfr[0] = bp[0];
      bfr[1] = bp[1];
      s = wmma4_f32(a, bfr, s);
    }
#pragma unroll
    for (int r = 0; r < 8; ++r)
      Ss[(w16 + fmb + r) * 148 + kt2 * 16 + fm] = s[r];
  }

  // ---- Band mask + scale + softmax (2 lanes per row, 72 cols each) ----
  {
    const float scale = 0.125f;  // 1/sqrt(64)
    const int row  = lane >> 1;          // query m (0..15)
    const int half = lane & 1;
    const int c0   = half * 72;
    const int jb   = kbase + w16;        // key j of col 0 for this wave
    float* srow = &Ss[(w16 + row) * 148];

    float mx = -INFINITY;
    for (int c = c0; c < c0 + 72; ++c) {
      bool ok = (jb + c >= 0) && (c >= row) && (c <= row + kWIN);
      float sv = ok ? srow[c] * scale : -INFINITY;
      mx = fmaxf(mx, sv);
    }
    mx = fmaxf(mx, __shfl_xor(mx, 1, 32));

    float sum = 0.0f;
    for (int c = c0; c < c0 + 72; ++c) {
      bool ok = (jb + c >= 0) && (c >= row) && (c <= row + kWIN);
      float e = ok ? __expf(srow[c] * scale - mx) : 0.0f;
      srow[c] = e;
      sum += e;
    }
    sum += __shfl_xor(sum, 1, 32);
    float inv = 1.0f / sum;
    for (int c = c0; c < c0 + 72; ++c) srow[c] *= inv;
  }

  // ---- O[16 x 64] = P(16x144) * V(144x64), 4 hd tiles x 36 k-slices ----
  float* outg = attn + (size_t)(b * kT + q0 + w16) * kD + h * kHD;
#pragma unroll
  for (int nt = 0; nt < 4; ++nt) {
    v8f o = v8f_zero();
    for (int kk = 0; kk < 36; ++kk) {  // 144 keys -> 36 slices of 4
      v2f a, bfr;
      const float* ap = &Ss[(w16 + fm) * 148 + kk * 4 + fk0];
      a[0] = ap[0];
      a[1] = ap[1];
      bfr[0] = Vs[(w16 + kk * 4 + fk0) * 72 + nt * 16 + fm];
      bfr[1] = Vs[(w16 + kk * 4 + fk0 + 1) * 72 + nt * 16 + fm];
      o = wmma4_f32(a, bfr, o);
    }
#pragma unroll
    for (int r = 0; r < 8; ++r)
      outg[(size_t)(fmb + r) * kD + nt * 16 + fm] = o[r];
  }
}

// ---------------------------------------------------------------------------
extern "C" void kernel_launch(void* const* d_in, const int* in_sizes, int n_in,
                              void* d_out, int out_size, void* d_ws,
                              size_t ws_size, hipStream_t stream) {
  (void)in_sizes; (void)n_in; (void)out_size; (void)ws_size;
  const float* x    = (const float*)d_in[0];  // [B,T,D]
  const float* Wqkv = (const float*)d_in[1];  // [D,3D]
  const float* Wout = (const float*)d_in[2];  // [D,D]
  float* out = (float*)d_out;                 // [B,T,D]

  float* qkv  = (float*)d_ws;                  // [8192, 2304]
  float* attn = qkv + (size_t)kM * kDq;        // [8192, 768]

  // 1) QKV projection: [8192,768] @ [768,2304]
  swa_gemm_f32_wmma<<<dim3(kDq / 64, kM / 128), 256, 0, stream>>>(
      x, Wqkv, qkv, kM, kDq, kD);

  // 2) Sliding-window attention: B*H*(T/32) blocks of 2 waves
  swa_attention_wmma<<<dim3(kB * kH * (kT / 32)), 64, 0, stream>>>(qkv, attn);

  // 3) Output projection: [8192,768] @ [768,768]
  swa_gemm_f32_wmma<<<dim3(kD / 64, kM / 128), 256, 0, stream>>>(
      attn, Wout, out, kM, kD, kD);
}